// spatial_transformer_89696097010069
// MI455X (gfx1250) — compile-verified
//
#include <hip/hip_runtime.h>

// Bilinear spatial-transformer sampler for MI455X (gfx1250).
//
// Memory-bound gather kernel: per output row, 4 gathered 256B-aligned 256B
// chunks of U + one streamed 256B output write.  U (128 MiB) fits the 192 MB
// global L2, so:
//   * U gathers use default RT temporal hint -> working set stays L2-resident;
//   * output (128 MiB, streamed, never re-read) uses the CDNA5 non-temporal
//     store hint so it does not evict U from L2;
//   * 16 consecutive lanes cover one 256B row chunk with global_load_b128
//     (100% cacheline utilization, wave32-friendly);
//   * 2 rows per thread -> 8 independent b128 gathers in flight;
//   * the 64 unique theta floats a block needs (2 x 128B, 16B-aligned) are
//     staged into LDS with one gfx1250 GLOBAL_LOAD_ASYNC_TO_LDS_B128 issued
//     by 16 lanes, synchronized via s_wait_asynccnt + workgroup barrier,
//     replacing 32 redundant broadcast global_load_b32 per block with cheap
//     ds_load reads.

typedef float v4f __attribute__((ext_vector_type(4)));

#define N_    8
#define C_    64
#define H_    256
#define W_    256
#define OH_   256
#define OW_   256
#define HWO_  (OH_ * OW_)           // 65536
#define ROWS_ (N_ * HWO_)           // 524288

__device__ __forceinline__ int clampi(int v, int lo, int hi) {
    v = v < lo ? lo : v;
    return v > hi ? hi : v;
}

__device__ __forceinline__ void row_params(float tx, float ty, int n, int p,
                                           size_t& ia, size_t& ib, size_t& ic, size_t& id,
                                           float& wa, float& wb, float& wc, float& wd) {
    const int ox = p & (OW_ - 1);
    const int oy = p >> 8;

    // identity grid: linspace(-1,1,256) -> step 2/255
    float x = tx + (-1.0f + (float)ox * (2.0f / (float)(OW_ - 1)));
    float y = ty + (-1.0f + (float)oy * (2.0f / (float)(OH_ - 1)));

    // map to pixel coordinates
    x = (x + 1.0f) * ((float)(W_ - 1) * 0.5f);
    y = (y + 1.0f) * ((float)(H_ - 1) * 0.5f);

    const int x0 = clampi((int)floorf(x), 0, W_ - 2);
    const int x1 = clampi((int)ceilf(x),  0, W_ - 1);
    const int y0 = clampi((int)floorf(y), 0, H_ - 2);
    const int y1 = clampi((int)ceilf(y),  0, H_ - 1);

    const int base = n * (H_ * W_);
    ia = (size_t)(unsigned)(base + y0 * W_ + x0) * C_;   // 256B-aligned float offsets
    ib = (size_t)(unsigned)(base + y1 * W_ + x0) * C_;
    ic = (size_t)(unsigned)(base + y0 * W_ + x1) * C_;
    id = (size_t)(unsigned)(base + y1 * W_ + x1) * C_;

    const float dx1 = (float)x1 - x;
    const float dx0 = x - (float)x0;
    const float dy1 = (float)y1 - y;
    const float dy0 = y - (float)y0;
    wa = dx1 * dy1;
    wb = dx1 * dy0;
    wc = dx0 * dy1;
    wd = dx0 * dy0;
}

__global__ __launch_bounds__(256) void st_bilinear_kernel(const float* __restrict__ U,
                                                          const float* __restrict__ theta,
                                                          float* __restrict__ out) {
    // s_th[0..31]  = theta_x for the block's 32 rows
    // s_th[32..63] = theta_y for the block's 32 rows
    __shared__ alignas(16) float s_th[64];

    const int tid      = threadIdx.x;
    const int blockRow = blockIdx.x * 32;
    const int n        = blockRow >> 16;          // image index (32 | 65536)
    const int p0       = blockRow & (HWO_ - 1);   // first pixel of block, 32-aligned

    // ---- async stage of theta into LDS (gfx1250 async-memory path) ----
    if (tid < 16) {
        const size_t thbase = (size_t)n * (2 * HWO_) + p0;    // 16B-aligned
        // lanes 0..7: x chunk (128B); lanes 8..15: y chunk (128B)
        const float* gsrc = theta + thbase + (tid < 8 ? tid * 4 : (HWO_ - 32) + tid * 4);
        const unsigned lds = (unsigned)(uintptr_t)s_th + (unsigned)tid * 16u;
        asm volatile("global_load_async_to_lds_b128 %0, %1, off"
                     :: "v"(lds), "v"(gsrc) : "memory");
    }
    asm volatile("s_wait_asynccnt 0" ::: "memory");   // no-op for non-issuing waves
    __syncthreads();                                  // publish LDS to all waves

    const int lane16 = tid & 15;        // lane within the 16-lane row group
    const int grp    = tid >> 4;        // row group within block (0..15)
    const int row0   = blockRow + grp;
    const int row1   = row0 + 16;
    const int coff   = lane16 * 4;      // this lane's float4 slice of 64 channels

    size_t ia0, ib0, ic0, id0, ia1, ib1, ic1, id1;
    float  wa0, wb0, wc0, wd0, wa1, wb1, wc1, wd1;
    row_params(s_th[grp],      s_th[32 + grp], n, p0 + grp,
               ia0, ib0, ic0, id0, wa0, wb0, wc0, wd0);
    row_params(s_th[16 + grp], s_th[48 + grp], n, p0 + grp + 16,
               ia1, ib1, ic1, id1, wa1, wb1, wc1, wd1);

    // Issue all 8 gathers (global_load_b128) before any use: deep MLP.
    const v4f A0 = *(const v4f*)(U + ia0 + coff);
    const v4f B0 = *(const v4f*)(U + ib0 + coff);
    const v4f C0 = *(const v4f*)(U + ic0 + coff);
    const v4f D0 = *(const v4f*)(U + id0 + coff);
    const v4f A1 = *(const v4f*)(U + ia1 + coff);
    const v4f B1 = *(const v4f*)(U + ib1 + coff);
    const v4f C1 = *(const v4f*)(U + ic1 + coff);
    const v4f D1 = *(const v4f*)(U + id1 + coff);

    v4f o0 = A0 * wa0;
    o0 += B0 * wb0;
    o0 += C0 * wc0;
    o0 += D0 * wd0;

    v4f o1 = A1 * wa1;
    o1 += B1 * wb1;
    o1 += C1 * wc1;
    o1 += D1 * wd1;

    // Streamed output: non-temporal so the 128 MiB write stream does not
    // evict the L2-resident U working set (CDNA5 TH cache-control).
    __builtin_nontemporal_store(o0, (v4f*)(out + (size_t)row0 * C_ + coff));
    __builtin_nontemporal_store(o1, (v4f*)(out + (size_t)row1 * C_ + coff));
}

extern "C" void kernel_launch(void* const* d_in, const int* in_sizes, int n_in,
                              void* d_out, int out_size, void* d_ws, size_t ws_size,
                              hipStream_t stream) {
    const float* U     = (const float*)d_in[0];
    const float* theta = (const float*)d_in[1];
    float*       out   = (float*)d_out;

    // 524288 rows, 32 rows per block (16 row groups x 2 rows per thread).
    const int blocks = ROWS_ / 32;      // 16384
    st_bilinear_kernel<<<blocks, 256, 0, stream>>>(U, theta, out);
}